// DeformRoIPooling_6468220748437
// MI455X (gfx1250) — compile-verified
//
#include <hip/hip_runtime.h>
#include <hip/hip_bf16.h>

// Problem constants (match reference setup_inputs)
#define BB 4
#define CC 256
#define HH 96
#define WW 96
#define HW (HH * WW)            // 9216
#define POOLED 7
#define NPARTS 7
#define NSAMP 4
#define SPATIAL_SCALE 0.0625f
#define TRANS_STD 0.1f

// ---- CDNA5 async global->LDS path (guarded; fallback compiles everywhere) ----
#ifndef USE_ASYNC_LDS
#if defined(__has_builtin)
#if __has_builtin(__builtin_amdgcn_global_load_async_to_lds_b32) && \
    __has_builtin(__builtin_amdgcn_s_wait_asynccnt)
#define USE_ASYNC_LDS 1
#endif
#endif
#endif
#ifndef USE_ASYNC_LDS
#define USE_ASYNC_LDS 0
#endif

// Parameter types per hipcc diagnostic: (addrspace(1) int*, addrspace(3) int*, imm, imm)
typedef __attribute__((address_space(1))) int* as1_i32p;
typedef __attribute__((address_space(3))) int* as3_i32p;

// =====================================================================
// Kernel 1: NCHW -> NHWC layout transform via LDS tile (64 spatial x 64 ch)
//   read  : coalesced along W, async global->LDS with transposed per-lane
//           scatter (ASYNCcnt-tracked, drained by s_wait_asynccnt)
//   LDS   : 64x65 pad -> conflict-free on both phases
//   write : coalesced along C
// =====================================================================
__global__ __launch_bounds__(256) void nchw_to_nhwc_kernel(
    const float* __restrict__ in, float* __restrict__ out) {
  __shared__ float tile[64 * 65];

  const int t   = threadIdx.x;
  const int col = t & 63;   // load: spatial-in-tile; store: channel-in-tile
  const int rq  = t >> 6;   // 0..3

  int blk = blockIdx.x;
  const int ctile = blk & 3;        blk >>= 2;   // 256/64 = 4 channel tiles
  const int stile = blk % 144;                   // 9216/64 = 144 spatial tiles
  const int b     = blk / 144;
  const int c0 = ctile * 64;
  const int s0 = stile * 64;

  const float* src = in + ((b * CC + c0) * HW) + s0;

#if USE_ASYNC_LDS
#pragma unroll
  for (int i = 0; i < 16; ++i) {
    const int r = i * 4 + rq;                       // channel row in tile
    const float* gp = src + r * HW + col;           // coalesced across lanes
    float* lp = &tile[col * 65 + r];                // transposed scatter
    __builtin_amdgcn_global_load_async_to_lds_b32((as1_i32p)gp, (as3_i32p)lp,
                                                  /*offset=*/0, /*cpol=*/0);
  }
  __builtin_amdgcn_s_wait_asynccnt(0);
  __syncthreads();
#else
#pragma unroll
  for (int i = 0; i < 16; ++i) {
    const int r = i * 4 + rq;
    tile[col * 65 + r] = src[r * HW + col];
  }
  __syncthreads();
#endif

  float* dst = out + ((b * HW + s0) * CC) + c0;
#pragma unroll
  for (int i = 0; i < 16; ++i) {
    const int srow = i * 4 + rq;                    // spatial row in tile
    dst[srow * CC + col] = tile[srow * 65 + col];   // coalesced across lanes
  }
}

// ---------------------------------------------------------------------
// Shared per-bin coordinate/weight math (all wave-uniform inputs).
// ---------------------------------------------------------------------
struct BinParams {
  int   b;
  float wstart, hstart, sub_w, sub_h;
};

__device__ __forceinline__ BinParams bin_params(const float* __restrict__ rois,
                                                const float* __restrict__ offs,
                                                int n, int ph, int pw) {
  BinParams bp;
  const float r0 = rois[n * 5 + 0];
  const float r1 = rois[n * 5 + 1];
  const float r2 = rois[n * 5 + 2];
  const float r3 = rois[n * 5 + 3];
  const float r4 = rois[n * 5 + 4];
  bp.b = (int)r0;

  const float rsw = rintf(r1) * SPATIAL_SCALE - 0.5f;  // jnp.round == rint (RNE)
  const float rsh = rintf(r2) * SPATIAL_SCALE - 0.5f;
  const float rew = rintf(r3 + 1.0f) * SPATIAL_SCALE - 0.5f;
  const float reh = rintf(r4 + 1.0f) * SPATIAL_SCALE - 0.5f;
  const float roi_w = fmaxf(rew - rsw, 0.1f);
  const float roi_h = fmaxf(reh - rsh, 0.1f);
  const float bin_w = roi_w / (float)POOLED;
  const float bin_h = roi_h / (float)POOLED;
  bp.sub_w = bin_w / (float)NSAMP;
  bp.sub_h = bin_h / (float)NSAMP;

  // replicate reference float sequence for part index
  const int part_h = (int)floorf((float)ph / (float)POOLED * (float)NPARTS);
  const int part_w = (int)floorf((float)pw / (float)POOLED * (float)NPARTS);
  const float tx =
      offs[((n * 2 + 0) * NPARTS + part_h) * NPARTS + part_w] * TRANS_STD;
  const float ty =
      offs[((n * 2 + 1) * NPARTS + part_h) * NPARTS + part_w] * TRANS_STD;

  bp.wstart = (float)pw * bin_w + rsw + tx * roi_w;
  bp.hstart = (float)ph * bin_h + rsh + ty * roi_h;
  return bp;
}

// =====================================================================
// Kernel 2a (NHWC fast path): 4 bins per 256-thread block; 64 lanes per
// bin, each lane owns 4 contiguous channels -> every corner gather is a
// wave-wide global_load_b128 (512B/instruction from L2-resident data).
// =====================================================================
__global__ __launch_bounds__(256) void deform_roi_pool_nhwc4_kernel(
    const float* __restrict__ x, const float* __restrict__ rois,
    const float* __restrict__ offs, float* __restrict__ out, int nbins) {
  const int group = threadIdx.x >> 6;          // bin-in-block, 0..3
  const int lane  = threadIdx.x & 63;          // channel quad id
  const int gbin  = blockIdx.x * 4 + group;
  if (gbin >= nbins) return;
  const int n   = gbin / (POOLED * POOLED);
  const int bin = gbin % (POOLED * POOLED);
  const int ph  = bin / POOLED;
  const int pw  = bin % POOLED;

  const BinParams bp = bin_params(rois, offs, n, ph, pw);

  float4 acc = make_float4(0.f, 0.f, 0.f, 0.f);
  int cnt = 0;

#pragma unroll
  for (int iy = 0; iy < NSAMP; ++iy) {
    const float hs = bp.hstart + (float)iy * bp.sub_h;
    const bool vy = (hs > -0.5f) && (hs < (float)HH - 0.5f);
    const float hc = fminf(fmaxf(hs, 0.0f), (float)(HH - 1));
    const int y0 = (int)floorf(hc);
    const int y1 = (int)ceilf(hc);
    const float dy = hc - (float)y0;
#pragma unroll
    for (int ix = 0; ix < NSAMP; ++ix) {
      const float ws = bp.wstart + (float)ix * bp.sub_w;
      if (!vy || !((ws > -0.5f) && (ws < (float)WW - 0.5f))) continue;  // uniform
      const float wc = fminf(fmaxf(ws, 0.0f), (float)(WW - 1));
      const int x0 = (int)floorf(wc);
      const int x1 = (int)ceilf(wc);
      const float dx = wc - (float)x0;

      const float4* p00 =
          (const float4*)(x + ((bp.b * HH + y0) * WW + x0) * CC);
      const float4* p01 =
          (const float4*)(x + ((bp.b * HH + y0) * WW + x1) * CC);
      const float4* p10 =
          (const float4*)(x + ((bp.b * HH + y1) * WW + x0) * CC);
      const float4* p11 =
          (const float4*)(x + ((bp.b * HH + y1) * WW + x1) * CC);
      const float4 v00 = p00[lane];   // global_load_b128, fully coalesced
      const float4 v01 = p01[lane];
      const float4 v10 = p10[lane];
      const float4 v11 = p11[lane];

      const float w00 = (1.0f - dx) * (1.0f - dy);
      const float w01 = dx * (1.0f - dy);
      const float w10 = (1.0f - dx) * dy;
      const float w11 = dx * dy;
      acc.x += w00 * v00.x + w01 * v01.x + w10 * v10.x + w11 * v11.x;
      acc.y += w00 * v00.y + w01 * v01.y + w10 * v10.y + w11 * v11.y;
      acc.z += w00 * v00.z + w01 * v01.z + w10 * v10.z + w11 * v11.z;
      acc.w += w00 * v00.w + w01 * v01.w + w10 * v10.w + w11 * v11.w;
      ++cnt;
    }
  }

  const float inv = (cnt > 0) ? (1.0f / (float)cnt) : 0.0f;
  const int c = lane * 4;
  const int obase = (n * CC + c) * (POOLED * POOLED) + ph * POOLED + pw;
  out[obase]                           = acc.x * inv;
  out[obase + 1 * (POOLED * POOLED)]   = acc.y * inv;
  out[obase + 2 * (POOLED * POOLED)]   = acc.z * inv;
  out[obase + 3 * (POOLED * POOLED)]   = acc.w * inv;
}

// =====================================================================
// Kernel 2b (NCHW fallback, used only if d_ws is too small): one bin per
// 256-thread block, thread = channel.
// =====================================================================
__global__ __launch_bounds__(256) void deform_roi_pool_nchw_kernel(
    const float* __restrict__ x, const float* __restrict__ rois,
    const float* __restrict__ offs, float* __restrict__ out, int N) {
  const int bin = blockIdx.x % (POOLED * POOLED);
  const int n   = blockIdx.x / (POOLED * POOLED);
  const int ph  = bin / POOLED;
  const int pw  = bin % POOLED;
  const int c   = threadIdx.x;
  if (n >= N) return;

  const BinParams bp = bin_params(rois, offs, n, ph, pw);

  float acc = 0.0f;
  int cnt = 0;

#pragma unroll
  for (int iy = 0; iy < NSAMP; ++iy) {
    const float hs = bp.hstart + (float)iy * bp.sub_h;
    const bool vy = (hs > -0.5f) && (hs < (float)HH - 0.5f);
    const float hc = fminf(fmaxf(hs, 0.0f), (float)(HH - 1));
    const int y0 = (int)floorf(hc);
    const int y1 = (int)ceilf(hc);
    const float dy = hc - (float)y0;
#pragma unroll
    for (int ix = 0; ix < NSAMP; ++ix) {
      const float ws = bp.wstart + (float)ix * bp.sub_w;
      if (!vy || !((ws > -0.5f) && (ws < (float)WW - 0.5f))) continue;
      const float wc = fminf(fmaxf(ws, 0.0f), (float)(WW - 1));
      const int x0 = (int)floorf(wc);
      const int x1 = (int)ceilf(wc);
      const float dx = wc - (float)x0;

      const float* p = x + (bp.b * CC + c) * HW;
      const float v00 = p[y0 * WW + x0];
      const float v01 = p[y0 * WW + x1];
      const float v10 = p[y1 * WW + x0];
      const float v11 = p[y1 * WW + x1];
      acc += (1.0f - dx) * (1.0f - dy) * v00 + dx * (1.0f - dy) * v01 +
             (1.0f - dx) * dy * v10 + dx * dy * v11;
      ++cnt;
    }
  }

  out[((n * CC + c) * POOLED + ph) * POOLED + pw] =
      (cnt > 0) ? (acc / (float)cnt) : 0.0f;
}

// =====================================================================
extern "C" void kernel_launch(void* const* d_in, const int* in_sizes, int n_in,
                              void* d_out, int out_size, void* d_ws,
                              size_t ws_size, hipStream_t stream) {
  (void)n_in;
  (void)out_size;
  const float* x    = (const float*)d_in[0];
  const float* rois = (const float*)d_in[1];
  const float* offs = (const float*)d_in[2];
  float* out = (float*)d_out;

  const int N = in_sizes[1] / 5;
  const int nbins = N * POOLED * POOLED;
  const size_t need = (size_t)BB * HW * CC * sizeof(float);  // 37.75 MB

  if (ws_size >= need) {
    float* xt = (float*)d_ws;
    nchw_to_nhwc_kernel<<<dim3(BB * 144 * 4), dim3(256), 0, stream>>>(x, xt);
    deform_roi_pool_nhwc4_kernel<<<dim3((nbins + 3) / 4), dim3(256), 0,
                                   stream>>>(xt, rois, offs, out, nbins);
  } else {
    deform_roi_pool_nchw_kernel<<<dim3(nbins), dim3(256), 0, stream>>>(
        x, rois, offs, out, N);
  }
}